// StochasticOpinion_79637283603163
// MI455X (gfx1250) — compile-verified
//
#include <hip/hip_runtime.h>

// Problem constants (from the reference): N=8192 points, D=64 dims.
#define NPTS 8192
#define DDIM 64
#define AUGW 68                 // 64 y-cols + sq + one + 2 zero-pad (K multiple of 4)
#define TILE 16
#define MROWS 32                // rows of output per wave (2 x 16-row WMMA tiles)
#define NJT (NPTS / TILE)       // 512 column tiles
#define NIB (NPTS / MROWS)      // 256 row groups

typedef float v2f __attribute__((ext_vector_type(2)));
typedef float v8f __attribute__((ext_vector_type(8)));

// ---------------------------------------------------------------------------
// Prep: build augmented matrices so one WMMA chain yields
//   d2 = ||yi||^2 + ||yj||^2 - 2*yi.yj   directly (K = 68),
// plus a column-major copy Yt[64][8192] for the second matmul's B frags.
// ---------------------------------------------------------------------------
__global__ void so_prep(const float* __restrict__ Y,
                        float* __restrict__ Aaug,
                        float* __restrict__ Baug,
                        float* __restrict__ Yt) {
  int i = blockIdx.x * blockDim.x + threadIdx.x;
  if (i >= NPTS) return;
  const float* yr = Y + (size_t)i * DDIM;
  float* ar = Aaug + (size_t)i * AUGW;
  float* br = Baug + (size_t)i * AUGW;
  float s = 0.f;
#pragma unroll 8
  for (int d = 0; d < DDIM; ++d) {
    float v = yr[d];
    s = fmaf(v, v, s);
    ar[d] = v;
    br[d] = -2.f * v;
    Yt[(size_t)d * NPTS + i] = v;      // coalesced across threads
  }
  ar[64] = s;   ar[65] = 1.f; ar[66] = 0.f; ar[67] = 0.f;
  br[64] = 1.f; br[65] = s;   br[66] = 0.f; br[67] = 0.f;
}

// ---------------------------------------------------------------------------
// Main fused kernel: one wave per (32-row group, j-slice); two independent
// 16x16 accumulator chains so every B fragment feeds two WMMAs and the
// WMMA RAW latency is hidden by interleaving.
// ---------------------------------------------------------------------------
__global__ __launch_bounds__(32) void so_main(
    const float* __restrict__ Aaug, const float* __restrict__ Baug,
    const float* __restrict__ Yt,
    const float* __restrict__ th1p, const float* __restrict__ th2p,
    float* __restrict__ phiY_part, float* __restrict__ rowsum_part,
    int tiles_per_slice) {
  __shared__ float lphi[2][TILE][TILE + 2];  // stride 18: b64-aligned, 0-conflict

  const int lane = threadIdx.x;   // 0..31 (wave32)
  const int half = lane >> 4;     // 0 or 1
  const int nn   = lane & 15;
  const int i0   = blockIdx.x * MROWS;
  const int js   = blockIdx.y;

  const float theta1 = *th1p;
  const float theta2 = *th2p;
  const float rcut   = theta2 + 1.0f;
  const float c2     = -0.01f * 1.44269504088896340736f;  // -0.01 * log2(e)
  const float inf    = __builtin_inff();

  // Loop-invariant A fragments for the two 16-row tiles.
  // f32 16x4 A layout: lanes 0-15 hold (K=4k,4k+1) of row M=lane,
  //                    lanes 16-31 hold (K=4k+2,4k+3) of row M=lane-16.
  v2f afrag0[17], afrag1[17];
#pragma unroll
  for (int k = 0; k < 17; ++k) {
    int kk = 4 * k + 2 * half;
    afrag0[k] = *(const v2f*)(Aaug + (size_t)(i0 + nn) * AUGW + kk);
    afrag1[k] = *(const v2f*)(Aaug + (size_t)(i0 + 16 + nn) * AUGW + kk);
  }

  v8f out0[4], out1[4];            // 32x64 phi@Y accumulator (4 n-chunks x 2)
#pragma unroll
  for (int c = 0; c < 4; ++c) { out0[c] = {}; out1[c] = {}; }
  v8f srow0 = {}, srow1 = {};      // phi row-sum accumulators (C layout)

  const int jt0 = js * tiles_per_slice;
  for (int jt = jt0; jt < jt0 + tiles_per_slice; ++jt) {
    const int j0 = jt * TILE;

    // ---- (1) two raw d2 tiles over K = 68, shared B fragments ----
    v8f g0 = {}, g1 = {};
#pragma unroll
    for (int k = 0; k < 17; ++k) {
      int kk = 4 * k + 2 * half;
      v2f b = *(const v2f*)(Baug + (size_t)(j0 + nn) * AUGW + kk);
      g0 = __builtin_amdgcn_wmma_f32_16x16x4_f32(false, afrag0[k], false, b,
                                                 (short)0, g0, false, false);
      g1 = __builtin_amdgcn_wmma_f32_16x16x4_f32(false, afrag1[k], false, b,
                                                 (short)0, g1, false, false);
    }

    // ---- (2) elementwise phi: med3 clamp + single-instruction TRANS ----
    float phiv0[8], phiv1[8];
#pragma unroll
    for (int v = 0; v < 8; ++v) {
      {
        float d2  = __builtin_amdgcn_fmed3f(g0[v], 0.f, inf);  // clamp >= 0
        float r   = __builtin_amdgcn_sqrtf(d2);                // v_sqrt_f32
        float dm  = r - theta2;
        float den = fmaf(dm, -dm, 1.f);                        // 1 - dm*dm
        float e   = c2 * __builtin_amdgcn_rcpf(den);           // v_rcp_f32
        float p   = theta1 * __builtin_amdgcn_exp2f(e);        // v_exp_f32
        p = (r >= rcut) ? 0.f : p;
        phiv0[v] = p;
        srow0[v] += p;
      }
      {
        float d2  = __builtin_amdgcn_fmed3f(g1[v], 0.f, inf);
        float r   = __builtin_amdgcn_sqrtf(d2);
        float dm  = r - theta2;
        float den = fmaf(dm, -dm, 1.f);
        float e   = c2 * __builtin_amdgcn_rcpf(den);
        float p   = theta1 * __builtin_amdgcn_exp2f(e);
        p = (r >= rcut) ? 0.f : p;
        phiv1[v] = p;
        srow1[v] += p;
      }
    }

    // ---- (3) C-layout -> A-layout transpose through LDS (both tiles) ----
#pragma unroll
    for (int v = 0; v < 8; ++v) {
      lphi[0][v + 8 * half][nn] = phiv0[v];
      lphi[1][v + 8 * half][nn] = phiv1[v];
    }
    __syncthreads();
    v2f pfrag0[4], pfrag1[4];
#pragma unroll
    for (int k2 = 0; k2 < 4; ++k2) {
      int kk = 4 * k2 + 2 * half;
      pfrag0[k2] = *(const v2f*)&lphi[0][nn][kk];            // ds_load_b64
      pfrag1[k2] = *(const v2f*)&lphi[1][nn][kk];
    }
    __syncthreads();

    // ---- (4) out += phi_tile @ Y_J, B frags shared across the two tiles ----
    // B frag lane (half,nn): Yt[(c*16+nn)*NPTS + j0+4k2+2*half + {0,1}]
#pragma unroll
    for (int c = 0; c < 4; ++c) {
      const float* bb = Yt + (size_t)(c * 16 + nn) * NPTS + j0 + 2 * half;
#pragma unroll
      for (int k2 = 0; k2 < 4; ++k2) {
        v2f b = *(const v2f*)(bb + 4 * k2);
        out0[c] = __builtin_amdgcn_wmma_f32_16x16x4_f32(
            false, pfrag0[k2], false, b, (short)0, out0[c], false, false);
        out1[c] = __builtin_amdgcn_wmma_f32_16x16x4_f32(
            false, pfrag1[k2], false, b, (short)0, out1[c], false, false);
      }
    }
  }

  // ---- write disjoint partials (full coverage -> no zero-init needed) ----
  float* pout = phiY_part + ((size_t)js * NPTS + i0) * DDIM;
#pragma unroll
  for (int c = 0; c < 4; ++c)
#pragma unroll
    for (int v = 0; v < 8; ++v) {
      pout[(size_t)(v + 8 * half) * DDIM + c * 16 + nn]      = out0[c][v];
      pout[(size_t)(16 + v + 8 * half) * DDIM + c * 16 + nn] = out1[c][v];
    }

  // reduce srow over the 16 lanes of each half (N dimension)
#pragma unroll
  for (int v = 0; v < 8; ++v) {
    float r0 = srow0[v];
    r0 += __shfl_xor(r0, 1, 32);
    r0 += __shfl_xor(r0, 2, 32);
    r0 += __shfl_xor(r0, 4, 32);
    r0 += __shfl_xor(r0, 8, 32);
    float r1 = srow1[v];
    r1 += __shfl_xor(r1, 1, 32);
    r1 += __shfl_xor(r1, 2, 32);
    r1 += __shfl_xor(r1, 4, 32);
    r1 += __shfl_xor(r1, 8, 32);
    if (nn == 0) {
      rowsum_part[(size_t)js * NPTS + i0 + v + 8 * half]      = r0;
      rowsum_part[(size_t)js * NPTS + i0 + 16 + v + 8 * half] = r1;
    }
  }
}

// ---------------------------------------------------------------------------
// Finalize: drift = phiY/N - y * (rowsum/N), fixed-order partial sum.
// ---------------------------------------------------------------------------
__global__ void so_finalize(const float* __restrict__ Y,
                            const float* __restrict__ phiY_part,
                            const float* __restrict__ rowsum_part,
                            float* __restrict__ out, int nslices) {
  int idx = blockIdx.x * blockDim.x + threadIdx.x;
  if (idx >= NPTS * DDIM) return;
  int i = idx >> 6;
  float acc = 0.f, rs = 0.f;
  for (int s = 0; s < nslices; ++s) {
    acc += phiY_part[(size_t)s * NPTS * DDIM + idx];
    rs  += rowsum_part[(size_t)s * NPTS + i];
  }
  const float invn = 1.0f / (float)NPTS;
  out[idx] = acc * invn - Y[idx] * (rs * invn);
}

// ---------------------------------------------------------------------------
extern "C" void kernel_launch(void* const* d_in, const int* in_sizes, int n_in,
                              void* d_out, int out_size, void* d_ws, size_t ws_size,
                              hipStream_t stream) {
  const float* Y   = (const float*)d_in[0];
  const float* th1 = (const float*)d_in[1];
  const float* th2 = (const float*)d_in[2];
  // d_in[3] = sigma (unused by the drift), d_in[4] = t (unused)
  float* out = (float*)d_out;

  // Workspace layout: [Aaug | Baug | Yt | phiY partials | rowsum partials]
  const size_t augBytes = (size_t)NPTS * AUGW * sizeof(float);
  const size_t ytBytes  = (size_t)NPTS * DDIM * sizeof(float);
  char* ws = (char*)d_ws;
  float* Aaug = (float*)ws;
  float* Baug = (float*)(ws + augBytes);
  float* Yt   = (float*)(ws + 2 * augBytes);
  char*  rest = ws + 2 * augBytes + ytBytes;
  size_t fixed = 2 * augBytes + ytBytes;
  size_t restBytes = (ws_size > fixed) ? (ws_size - fixed) : 0;
  const size_t perSlice = ((size_t)NPTS * DDIM + NPTS) * sizeof(float);

  int nslices = 1;                          // must divide NJT (512)
  if (restBytes >= 8 * perSlice)      nslices = 8;
  else if (restBytes >= 4 * perSlice) nslices = 4;
  else if (restBytes >= 2 * perSlice) nslices = 2;

  float* phiY_part   = (float*)rest;
  float* rowsum_part = (float*)(rest + (size_t)nslices * NPTS * DDIM * sizeof(float));

  so_prep<<<dim3((NPTS + 255) / 256), dim3(256), 0, stream>>>(Y, Aaug, Baug, Yt);

  so_main<<<dim3(NIB, nslices), dim3(32), 0, stream>>>(
      Aaug, Baug, Yt, th1, th2, phiY_part, rowsum_part, NJT / nslices);

  so_finalize<<<dim3((NPTS * DDIM + 255) / 256), dim3(256), 0, stream>>>(
      Y, phiY_part, rowsum_part, out, nslices);
}